// Dynamics_69106023793032
// MI455X (gfx1250) — compile-verified
//
#include <hip/hip_runtime.h>

typedef float v2f __attribute__((ext_vector_type(2)));
typedef float v8f __attribute__((ext_vector_type(8)));

#define NLAYERS 10
#define NFEAT   256

// ---------------------------------------------------------------------------
// Prep: compute a = f(attractor) once (trivial scalar work, ~2560 exps).
// ---------------------------------------------------------------------------
__global__ void diffeo_attractor_prep(const float* __restrict__ centers,
                                      const float* __restrict__ Ws,
                                      const float* __restrict__ Wt,
                                      const float* __restrict__ att,
                                      const float* __restrict__ sigma,
                                      float* __restrict__ aout) {
  if (threadIdx.x != 0 || blockIdx.x != 0) return;
  float s = sigma[0];
  float inv2s2 = 1.0f / (2.0f * s * s);
  float x0 = att[0], x1 = att[1];
  for (int l = 0; l < NLAYERS; ++l) {
    int ia = l & 1;                      // conditioning coordinate index
    float xa = ia ? x1 : x0;
    float xb = ia ? x0 : x1;
    const float* cl  = centers + l * NFEAT;
    const float* wsl = Ws + l * NFEAT;
    const float* wtl = Wt + l * NFEAT;
    float S = 0.f, T = 0.f;
    for (int f = 0; f < NFEAT; ++f) {
      float d = xa - cl[f];
      float k = __expf(-d * d * inv2s2);
      S += k * wsl[f];
      T += k * wtl[f];
    }
    float nb = xb * __expf(S) + T;
    if (ia) x0 = nb; else x1 = nb;
  }
  aout[0] = x0;
  aout[1] = x1;
}

// ---------------------------------------------------------------------------
// Main kernel: each wave32 owns a tile of 16 points.
//  - A-matrix (16 pts x 4 feats, f32): lane L holds point (L&15); lanes 0-15
//    supply K=0,1 and lanes 16-31 supply K=2,3 (ISA 16x4 f32 A layout).
//  - B-matrix (4 feats x 16 cols): cols 0..3 = [Ws, Wt, c*Ws, c*Wt], rest 0.
//  - 64 chained v_wmma_f32_16x16x4_f32 per layer accumulate S,T,SC,TC.
//  - Jacobian propagated in closed form (forward mode, 2x2 per point).
// ---------------------------------------------------------------------------
__global__ __launch_bounds__(256) void diffeo_dyn_kernel(
    const float* __restrict__ x,
    const float* __restrict__ centers,
    const float* __restrict__ Ws,
    const float* __restrict__ Wt,
    const float* __restrict__ sigma,
    const float* __restrict__ avec,
    float* __restrict__ out,
    int M) {
  __shared__ float centS[NLAYERS * NFEAT];        // 10 KB
  __shared__ float bpS[NLAYERS * NFEAT * 4];      // 40 KB: Ws,Wt,c*Ws,c*Wt
  __shared__ float scr[8][16][4];                 // per-wave transpose scratch

  const int tid = threadIdx.x;

  // Stage weights into LDS (uniform across all tiles).
  for (int i = tid; i < NLAYERS * NFEAT; i += 256) {
    float c  = centers[i];
    float ws = Ws[i];
    float wt = Wt[i];
    centS[i]      = c;
    bpS[i * 4 + 0] = ws;
    bpS[i * 4 + 1] = wt;
    bpS[i * 4 + 2] = c * ws;
    bpS[i * 4 + 3] = c * wt;
  }
  __syncthreads();

  const float sg = sigma[0];
  const float inv2s2 = 1.0f / (2.0f * sg * sg);
  const float a0 = avec[0], a1 = avec[1];

  const int wave = tid >> 5;
  const int lane = tid & 31;
  const int hi   = lane >> 4;        // which half of the wave
  const int col  = lane & 15;        // point index within tile / D column
  const int colm = col & 3;
  const bool colOK = (col < 4);
  const int foff = hi * 2;           // feature offset for this half (K=2,3)

  const int tile = blockIdx.x * 8 + wave;
  int p = tile * 16 + col;
  int pc = (p < M) ? p : (M - 1);    // clamp (grid is exact for M=262144)

  float x0 = x[2 * pc], x1 = x[2 * pc + 1];
  float j00 = 1.f, j01 = 0.f, j10 = 0.f, j11 = 1.f;   // J[k][i] = dy_k/dx_i

  for (int l = 0; l < NLAYERS; ++l) {
    const int ia = l & 1;
    const float xa = ia ? x1 : x0;
    const float xb = ia ? x0 : x1;
    const float* cl = &centS[l * NFEAT];
    const float* bl = &bpS[l * NFEAT * 4];

    v8f acc = {0.f, 0.f, 0.f, 0.f, 0.f, 0.f, 0.f, 0.f};
    #pragma unroll 4
    for (int i2 = 0; i2 < 64; ++i2) {
      const int f = i2 * 4 + foff;
      float c0 = cl[f];
      float c1 = cl[f + 1];
      float d0 = xa - c0;
      float d1 = xa - c1;
      v2f A;
      A.x = __expf(-d0 * d0 * inv2s2);
      A.y = __expf(-d1 * d1 * inv2s2);
      float b0 = bl[f * 4 + colm];
      float b1 = bl[(f + 1) * 4 + colm];
      v2f B;
      B.x = colOK ? b0 : 0.f;
      B.y = colOK ? b1 : 0.f;
      acc = __builtin_amdgcn_wmma_f32_16x16x4_f32(
          false, A, false, B, (short)0, acc, false, false);
    }

    // Transpose D cols 0..3 through LDS so lane L gets point (L&15)'s sums.
    if (colOK) {
      #pragma unroll
      for (int r = 0; r < 8; ++r) scr[wave][r + 8 * hi][colm] = acc[r];
    }
    __builtin_amdgcn_wave_barrier();
    float S  = scr[wave][col][0];
    float T  = scr[wave][col][1];
    float SC = scr[wave][col][2];
    float TC = scr[wave][col][3];
    __builtin_amdgcn_wave_barrier();

    // ds/dxa = -(xa*S - SC)/sigma^2 ; same for t.
    float sp = -2.f * inv2s2 * (xa * S - SC);
    float tp = -2.f * inv2s2 * (xa * T - TC);
    float es = __expf(S);
    float coef = xb * es * sp + tp;
    float nb = xb * es + T;
    if (ia == 0) {            // xa = coord0, warp coord1: update J row 1
      j10 = es * j10 + coef * j00;
      j11 = es * j11 + coef * j01;
      x1 = nb;
    } else {                  // xa = coord1, warp coord0: update J row 0
      j00 = es * j00 + coef * j10;
      j01 = es * j01 + coef * j11;
      x0 = nb;
    }
  }

  // G = J^T J ; out = -G^{-1} (y - a)
  float g00 = j00 * j00 + j10 * j10;
  float g01 = j00 * j01 + j10 * j11;
  float g11 = j01 * j01 + j11 * j11;
  float det = g00 * g11 - g01 * g01;
  float invd = 1.0f / det;
  float d0 = x0 - a0, d1 = x1 - a1;
  float o0 = -(g11 * d0 - g01 * d1) * invd;
  float o1 = -(g00 * d1 - g01 * d0) * invd;

  if (hi == 0 && p < M) {
    out[2 * p]     = o0;
    out[2 * p + 1] = o1;
  }
}

// ---------------------------------------------------------------------------
extern "C" void kernel_launch(void* const* d_in, const int* in_sizes, int n_in,
                              void* d_out, int out_size, void* d_ws, size_t ws_size,
                              hipStream_t stream) {
  (void)n_in; (void)out_size; (void)ws_size;
  const float* x       = (const float*)d_in[0];
  const float* centers = (const float*)d_in[1];
  const float* Ws      = (const float*)d_in[2];
  const float* Wt      = (const float*)d_in[3];
  const float* att     = (const float*)d_in[4];
  const float* sigma   = (const float*)d_in[5];
  float* out = (float*)d_out;
  float* aws = (float*)d_ws;   // 2 floats: f(attractor)

  const int M = in_sizes[0] / 2;

  diffeo_attractor_prep<<<1, 32, 0, stream>>>(centers, Ws, Wt, att, sigma, aws);

  const int tiles  = (M + 15) / 16;       // 16 points per wave
  const int blocks = (tiles + 7) / 8;     // 8 waves per 256-thread block
  diffeo_dyn_kernel<<<blocks, 256, 0, stream>>>(x, centers, Ws, Wt, sigma,
                                                aws, out, M);
}